// NEST_TA_24489903522481
// MI455X (gfx1250) — compile-verified
//
#include <hip/hip_runtime.h>
#include <math.h>

// MI455X / gfx1250, wave32. fp32 end-to-end (problem is L2-bandwidth bound:
// 16MB Struct resident in 192MB L2; ~0.3 GFLOP total). Matrix work done with
// V_WMMA_F32_16X16X4_F32 Gram products per 16-row tile.

typedef __attribute__((ext_vector_type(2))) float v2f;
typedef __attribute__((ext_vector_type(8))) float v8f;

#define B_ROWS 8192
#define D_COLS 512
#define KREF   8

#if defined(__has_builtin)
#if __has_builtin(__builtin_amdgcn_wmma_f32_16x16x4_f32)
#define HAVE_WMMA_F32X4 1
#endif
#endif

// ---------------------------------------------------------------------------
// Kernel 1: per-row top-8 nearest labels + normalized Gaussian weights.
// One thread per row; all 8192 labels staged in LDS (32 KB).
// ---------------------------------------------------------------------------
__global__ __launch_bounds__(256) void topk_weights_kernel(
    const float* __restrict__ Label,
    float* __restrict__ wOut,
    int* __restrict__ idxOut) {
  __shared__ float lab[B_ROWS];
  const int tid = threadIdx.x;
  for (int j = tid; j < B_ROWS; j += 256) lab[j] = Label[j];
  __syncthreads();

  const int i = blockIdx.x * 256 + tid;
  const float Li = lab[i];

  float bd[KREF];
  int bi[KREF];
#pragma unroll
  for (int k = 0; k < KREF; ++k) { bd[k] = 3.4e38f; bi[k] = 0; }

  const float4* lab4 = (const float4*)lab;
  for (int j4 = 0; j4 < B_ROWS / 4; ++j4) {
    const float4 v = lab4[j4];
    float dv[4];
    dv[0] = fabsf(v.x - Li);
    dv[1] = fabsf(v.y - Li);
    dv[2] = fabsf(v.z - Li);
    dv[3] = fabsf(v.w - Li);
#pragma unroll
    for (int c = 0; c < 4; ++c) {
      float dist = dv[c];
      if (dist < bd[KREF - 1]) {          // rare after warm-up
        int id = j4 * 4 + c;
#pragma unroll
        for (int p = 0; p < KREF; ++p) {  // stable insertion (strict <):
          if (dist < bd[p]) {             // ties keep earlier index, like JAX
            float td = bd[p]; bd[p] = dist; dist = td;
            int   ti = bi[p]; bi[p] = id;  id  = ti;
          }
        }
      }
    }
  }

  // Gaussian weights; the 1/(sqrt(2*pi)*STD) constant cancels in the
  // normalization.  exp(-(d^2)/(2*25)) = exp(-d^2 * 0.02)
  float wv[KREF];
  float s = 0.f;
#pragma unroll
  for (int k = 0; k < KREF; ++k) {
    wv[k] = expf(-bd[k] * bd[k] * 0.02f);
    s += wv[k];
  }
  const float inv = 1.f / s;  // self-match guarantees s >= 1
#pragma unroll
  for (int k = 0; k < KREF; ++k) {
    wOut[i * KREF + k]   = wv[k] * inv;
    idxOut[i * KREF + k] = bi[k];
  }
}

// ---------------------------------------------------------------------------
// Kernel 2: one wave per 16-row tile. Builds S and Sm fragments in the fp32
// WMMA A/B layout and accumulates three 16x16 Grams with
// v_wmma_f32_16x16x4_f32; diagonals give S.Sm, |S|^2, |Sm|^2 per row.
// ---------------------------------------------------------------------------
__device__ __forceinline__ float pick8(v8f v, int k) {
  float r = v[0];
  r = (k == 1) ? v[1] : r;
  r = (k == 2) ? v[2] : r;
  r = (k == 3) ? v[3] : r;
  r = (k == 4) ? v[4] : r;
  r = (k == 5) ? v[5] : r;
  r = (k == 6) ? v[6] : r;
  r = (k == 7) ? v[7] : r;
  return r;
}

__global__ __launch_bounds__(32) void sim_tile_kernel(
    const float* __restrict__ S,
    const float* __restrict__ wAll,
    const int* __restrict__ idxAll,
    float* __restrict__ sims) {
  const int lane = threadIdx.x;        // 0..31, wave32
  const int r0   = blockIdx.x * 16;    // tile of 16 rows
  const int m    = lane & 15;          // my matrix row
  const int half = lane >> 4;          // 0: K0..1, 1: K2..3 (fp32 A/B layout)
  const int row  = r0 + m;

  float w[KREF];
  int   ridx[KREF];
#pragma unroll
  for (int t = 0; t < KREF; ++t) {
    w[t]    = wAll[row * KREF + t];
    ridx[t] = idxAll[row * KREF + t];
  }

  const v2f* Srow = (const v2f*)(S + row * D_COLS);

#ifdef HAVE_WMMA_F32X4
  v8f accSSm = {};  // diag -> S_b . Sm_b
  v8f accSS  = {};  // diag -> |S_b|^2
  v8f accMM  = {};  // diag -> |Sm_b|^2
#else
  float pSSm = 0.f, pSS = 0.f, pMM = 0.f;
#endif

  for (int d = 0; d < D_COLS; d += 4) {
    const int c2 = (d >> 1) + half;     // float2 index: cols d+2*half, +1
    const v2f a = Srow[c2];             // S fragment (A layout)

    v2f sm = {0.f, 0.f};                // Sm fragment (same layout -> B)
#pragma unroll
    for (int t = 0; t < KREF; ++t) {
      const v2f g = ((const v2f*)(S + ridx[t] * D_COLS))[c2];
      sm += w[t] * g;
    }

#ifdef HAVE_WMMA_F32X4
    // fp32 A 16x4 layout == fp32 B 4x16 layout with M<->N, so feeding an
    // "A-layout" fragment as SRC1 computes X * Y^T.
    accSSm = __builtin_amdgcn_wmma_f32_16x16x4_f32(
        false, a, false, sm, (short)0, accSSm, false, false);
    accSS = __builtin_amdgcn_wmma_f32_16x16x4_f32(
        false, a, false, a, (short)0, accSS, false, false);
    accMM = __builtin_amdgcn_wmma_f32_16x16x4_f32(
        false, sm, false, sm, (short)0, accMM, false, false);
#else
    pSSm += a.x * sm.x + a.y * sm.y;
    pSS  += a.x * a.x + a.y * a.y;
    pMM  += sm.x * sm.x + sm.y * sm.y;
#endif
  }

#ifdef HAVE_WMMA_F32X4
  // C/D layout: VGPR v holds row M=v (lanes 0-15, N=lane) and row M=v+8
  // (lanes 16-31, N=lane-16).  Diagonal (m,m): m<8 -> lane m, acc[m];
  // m>=8 -> lane m+16, acc[m-8].
  const bool lo = (lane < 8);
  const bool hi = (lane >= 24);
  const int  k  = lo ? lane : (lane - 24);
  const int  dr = lo ? lane : (lane - 16);
  if (lo || hi) {
    const float dss = pick8(accSS, k);
    const float dsm = pick8(accSSm, k);
    const float dmm = pick8(accMM, k);
    const float sim =
        dsm / ((1e-10f + sqrtf(dss)) * (1e-10f + sqrtf(dmm)));
    sims[r0 + dr] = sim;
  }
#else
  // Fallback: combine the two half-lanes holding the same row.
  const float oSSm = pSSm + __shfl_xor(pSSm, 16, 32);
  const float oSS  = pSS  + __shfl_xor(pSS, 16, 32);
  const float oMM  = pMM  + __shfl_xor(pMM, 16, 32);
  if (lane < 16) {
    sims[row] = oSSm / ((1e-10f + sqrtf(oSS)) * (1e-10f + sqrtf(oMM)));
  }
#endif
}

// ---------------------------------------------------------------------------
// Kernel 3: deterministic tree reduction -> out[0] = 1 - mean(sims)
// ---------------------------------------------------------------------------
__global__ __launch_bounds__(256) void reduce_kernel(
    const float* __restrict__ sims, float* __restrict__ out) {
  __shared__ float buf[256];
  const int tid = threadIdx.x;
  float s = 0.f;
  for (int j = tid; j < B_ROWS; j += 256) s += sims[j];
  buf[tid] = s;
  __syncthreads();
  for (int off = 128; off > 0; off >>= 1) {
    if (tid < off) buf[tid] += buf[tid + off];
    __syncthreads();
  }
  if (tid == 0) out[0] = 1.0f - buf[0] * (1.0f / (float)B_ROWS);
}

// ---------------------------------------------------------------------------
extern "C" void kernel_launch(void* const* d_in, const int* in_sizes, int n_in,
                              void* d_out, int out_size, void* d_ws,
                              size_t ws_size, hipStream_t stream) {
  (void)in_sizes; (void)n_in; (void)out_size; (void)ws_size;
  const float* Struct = (const float*)d_in[0];  // [8192, 512] f32
  const float* Label  = (const float*)d_in[1];  // [8192] f32
  float* out = (float*)d_out;                   // scalar f32

  char* ws = (char*)d_ws;
  float* wBuf   = (float*)ws;                                     // 256 KB
  int*   idxBuf = (int*)(ws + (size_t)B_ROWS * KREF * 4);         // 256 KB
  float* sims   = (float*)(ws + 2 * (size_t)B_ROWS * KREF * 4);   //  32 KB

  topk_weights_kernel<<<B_ROWS / 256, 256, 0, stream>>>(Label, wBuf, idxBuf);
  sim_tile_kernel<<<B_ROWS / 16, 32, 0, stream>>>(Struct, wBuf, idxBuf, sims);
  reduce_kernel<<<1, 256, 0, stream>>>(sims, out);
}